// GraphAttentionLayer_43576738185629
// MI455X (gfx1250) — compile-verified
//
#include <hip/hip_runtime.h>
#include <math.h>

// GAT layer: h = xW; per-head logits; edge segment-softmax; scatter-agg;
// residual + exact GELU + LayerNorm.  fp32 throughout (matches reference).

#define DIM       128
#define HEADS     4
#define HEAD_C    32
#define NEG_SLOPE 0.2f
#define LN_EPS    1e-5f

typedef __attribute__((ext_vector_type(2))) float v2f;
typedef __attribute__((ext_vector_type(8))) float v8f;

// ---------------------------------------------------------------------------
// Kernel 1: h = x @ W via V_WMMA_F32_16X16X4_F32.
// Block = 256 threads = 8 waves. Each wave computes a 16-row x 128-col strip.
// W (64 KB) is staged into LDS once per block and shared by all 8 waves.
// A-frag (16x4):  lane<16 -> row=lane, v0=K0,v1=K1 ; lane>=16 -> v0=K2,v1=K3
// B-frag (4x16):  lane%16 = col, v0=K(koff), v1=K(koff+1), koff=2*(lane/16)
// C/D (16x16):    vgpr v: lanes0-15 -> M=v, lanes16-31 -> M=v+8; N=lane%16
// ---------------------------------------------------------------------------
__global__ __launch_bounds__(256) void gat_gemm_wmma(
    const float* __restrict__ x, const float* __restrict__ W,
    float* __restrict__ h, int nNodes) {
  __shared__ float sW[DIM * DIM];  // 64 KB of 320 KB WGP LDS
  {
    const float4* Wv = (const float4*)W;
    float4* sWv = (float4*)sW;
    for (int i = threadIdx.x; i < DIM * DIM / 4; i += 256) sWv[i] = Wv[i];
  }
  __syncthreads();

  const int wave = threadIdx.x >> 5;
  const int lane = threadIdx.x & 31;
  const int strip = blockIdx.x * 8 + wave;          // 16-row strip index
  if (strip * 16 >= nNodes) return;                 // uniform per wave; EXEC stays all-1s

  const int lrow = lane & 15;
  const int koff = (lane >> 4) << 1;                // 0 or 2
  const float* __restrict__ xrow = x + (size_t)(strip * 16 + lrow) * DIM;

  v8f c[8] = {};
  for (int kb = 0; kb < DIM; kb += 4) {
    v2f a;
    a.x = xrow[kb + koff];
    a.y = xrow[kb + koff + 1];
#pragma unroll
    for (int t = 0; t < 8; ++t) {
      v2f bm;
      const int col = t * 16 + lrow;
      bm.x = sW[(kb + koff) * DIM + col];
      bm.y = sW[(kb + koff + 1) * DIM + col];
      c[t] = __builtin_amdgcn_wmma_f32_16x16x4_f32(
          /*neg_a=*/false, a, /*neg_b=*/false, bm,
          /*c_mod=*/(short)0, c[t], /*reuse_a=*/false, /*reuse_b=*/false);
    }
  }

  const int mbase = strip * 16 + ((lane >> 4) << 3);
#pragma unroll
  for (int t = 0; t < 8; ++t) {
#pragma unroll
    for (int v = 0; v < 8; ++v) {
      h[(size_t)(mbase + v) * DIM + t * 16 + lrow] = c[t][v];
    }
  }
}

// ---------------------------------------------------------------------------
// Kernel 2: init scratch (agg=0, m=-inf, denom=0)
// ---------------------------------------------------------------------------
__global__ void gat_init(float* __restrict__ agg, float* __restrict__ m,
                         float* __restrict__ denom, int nNodes) {
  const int total = nNodes * DIM;
  const int nh = nNodes * HEADS;
  for (int i = blockIdx.x * blockDim.x + threadIdx.x; i < total;
       i += gridDim.x * blockDim.x) {
    agg[i] = 0.0f;
    if (i < nh) { m[i] = -3.0e38f; denom[i] = 0.0f; }
  }
}

// ---------------------------------------------------------------------------
// Kernel 3: per-node attention logits. One block per node, 128 threads =
// 4 wave32s = 4 heads; wave-shuffle reduction over the 32 head channels.
// ---------------------------------------------------------------------------
__global__ __launch_bounds__(128) void gat_att(
    const float* __restrict__ h, const float* __restrict__ att_src,
    const float* __restrict__ att_dst, float* __restrict__ a_s,
    float* __restrict__ a_d) {
  const int n = blockIdx.x;
  const int t = threadIdx.x;
  const float hv = h[(size_t)n * DIM + t];
  float s = hv * att_src[t];
  float d = hv * att_dst[t];
#pragma unroll
  for (int off = 16; off > 0; off >>= 1) {
    s += __shfl_down(s, off, 32);
    d += __shfl_down(d, off, 32);
  }
  if ((t & 31) == 0) {
    a_s[n * HEADS + (t >> 5)] = s;
    a_d[n * HEADS + (t >> 5)] = d;
  }
}

__device__ __forceinline__ void edge_src_dst(const int* __restrict__ ei, int E,
                                             int e, int& src, int& dst) {
  if (e < E) { src = ei[e]; dst = ei[E + e]; }
  else       { src = dst = e - E; }   // self loop
}

// ---------------------------------------------------------------------------
// Kernel 4: segment max -> GLOBAL_ATOMIC_MAX_NUM_F32
// ---------------------------------------------------------------------------
__global__ void gat_edge_max(const int* __restrict__ ei, int E, int Etot,
                             const float* __restrict__ a_s,
                             const float* __restrict__ a_d,
                             float* __restrict__ m) {
  const long long idx = (long long)blockIdx.x * blockDim.x + threadIdx.x;
  if (idx >= (long long)Etot * HEADS) return;
  const int e = (int)(idx >> 2);
  const int hd = (int)(idx & 3);
  int src, dst;
  edge_src_dst(ei, E, e, src, dst);
  float v = a_s[src * HEADS + hd] + a_d[dst * HEADS + hd];
  v = v > 0.0f ? v : NEG_SLOPE * v;
  atomicMax(&m[dst * HEADS + hd], v);
}

// ---------------------------------------------------------------------------
// Kernel 5: denom = segment_sum(exp(e - m[dst])) -> GLOBAL_ATOMIC_ADD_F32
// ---------------------------------------------------------------------------
__global__ void gat_edge_denom(const int* __restrict__ ei, int E, int Etot,
                               const float* __restrict__ a_s,
                               const float* __restrict__ a_d,
                               const float* __restrict__ m,
                               float* __restrict__ denom) {
  const long long idx = (long long)blockIdx.x * blockDim.x + threadIdx.x;
  if (idx >= (long long)Etot * HEADS) return;
  const int e = (int)(idx >> 2);
  const int hd = (int)(idx & 3);
  int src, dst;
  edge_src_dst(ei, E, e, src, dst);
  float v = a_s[src * HEADS + hd] + a_d[dst * HEADS + hd];
  v = v > 0.0f ? v : NEG_SLOPE * v;
  atomicAdd(&denom[dst * HEADS + hd], __expf(v - m[dst * HEADS + hd]));
}

// ---------------------------------------------------------------------------
// Kernel 6: weighted scatter-aggregation. One block (128 threads) per edge;
// thread t handles channel t (head = t/32). alpha recomputed per thread to
// avoid a 10 MB ez buffer; atomic f32 adds into agg[dst].
// ---------------------------------------------------------------------------
__global__ __launch_bounds__(128) void gat_scatter(
    const int* __restrict__ ei, int E,
    const float* __restrict__ a_s, const float* __restrict__ a_d,
    const float* __restrict__ m, const float* __restrict__ denom,
    const float* __restrict__ h, float* __restrict__ agg) {
  const int e = blockIdx.x;
  const int t = threadIdx.x;
  const int hd = t >> 5;
  int src, dst;
  edge_src_dst(ei, E, e, src, dst);
  float v = a_s[src * HEADS + hd] + a_d[dst * HEADS + hd];
  v = v > 0.0f ? v : NEG_SLOPE * v;
  const float alpha =
      __expf(v - m[dst * HEADS + hd]) / (denom[dst * HEADS + hd] + 1e-16f);
  atomicAdd(&agg[(size_t)dst * DIM + t], alpha * h[(size_t)src * DIM + t]);
}

// ---------------------------------------------------------------------------
// Kernel 7: residual + exact-erf GELU + LayerNorm. One block per node.
// ---------------------------------------------------------------------------
__global__ __launch_bounds__(128) void gat_epilogue(
    const float* __restrict__ x, const float* __restrict__ b,
    const float* __restrict__ gamma, const float* __restrict__ beta,
    const float* __restrict__ agg, float* __restrict__ out) {
  __shared__ float red[8];
  const int n = blockIdx.x;
  const int t = threadIdx.x;
  const float g = agg[(size_t)n * DIM + t] + b[t];
  const float ge = 0.5f * g * (1.0f + erff(g * 0.70710678118654752f));
  const float y = x[(size_t)n * DIM + t] + ge;

  float s = y;
#pragma unroll
  for (int off = 16; off > 0; off >>= 1) s += __shfl_down(s, off, 32);
  if ((t & 31) == 0) red[t >> 5] = s;
  __syncthreads();
  const float mu = (red[0] + red[1] + red[2] + red[3]) * (1.0f / DIM);

  const float d = y - mu;
  float s2 = d * d;
#pragma unroll
  for (int off = 16; off > 0; off >>= 1) s2 += __shfl_down(s2, off, 32);
  if ((t & 31) == 0) red[4 + (t >> 5)] = s2;
  __syncthreads();
  const float var = (red[4] + red[5] + red[6] + red[7]) * (1.0f / DIM);

  out[(size_t)n * DIM + t] = d * rsqrtf(var + LN_EPS) * gamma[t] + beta[t];
}

// ---------------------------------------------------------------------------
extern "C" void kernel_launch(void* const* d_in, const int* in_sizes, int n_in,
                              void* d_out, int out_size, void* d_ws,
                              size_t ws_size, hipStream_t stream) {
  const float* x       = (const float*)d_in[0];
  const float* W       = (const float*)d_in[1];
  const float* b       = (const float*)d_in[2];
  const float* att_src = (const float*)d_in[3];
  const float* att_dst = (const float*)d_in[4];
  const float* gamma   = (const float*)d_in[5];
  const float* beta    = (const float*)d_in[6];
  const int*   ei      = (const int*)d_in[7];

  const int nNodes = in_sizes[0] / DIM;   // 50000 (multiple of 16)
  const int E      = in_sizes[7] / 2;     // 600000
  const int Etot   = E + nNodes;          // + self loops

  // Workspace layout (256B aligned): h, agg, a_s, a_d, m, denom  (~54 MB)
  char* ws = (char*)d_ws;
  size_t off = 0;
  auto alloc = [&](size_t bytes) -> float* {
    float* p = (float*)(ws + off);
    off = (off + bytes + 255) & ~(size_t)255;
    return p;
  };
  float* h     = alloc((size_t)nNodes * DIM * sizeof(float));
  float* agg   = alloc((size_t)nNodes * DIM * sizeof(float));
  float* a_s   = alloc((size_t)nNodes * HEADS * sizeof(float));
  float* a_d   = alloc((size_t)nNodes * HEADS * sizeof(float));
  float* m     = alloc((size_t)nNodes * HEADS * sizeof(float));
  float* denom = alloc((size_t)nNodes * HEADS * sizeof(float));
  (void)ws_size; (void)n_in; (void)out_size;

  gat_init<<<2048, 256, 0, stream>>>(agg, m, denom, nNodes);

  const int nStrips = nNodes / 16;                     // 3125
  gat_gemm_wmma<<<(nStrips + 7) / 8, 256, 0, stream>>>(x, W, h, nNodes);

  gat_att<<<nNodes, 128, 0, stream>>>(h, att_src, att_dst, a_s, a_d);

  const long long eh = (long long)Etot * HEADS;
  const int eblk = 256;
  const int egrid = (int)((eh + eblk - 1) / eblk);
  gat_edge_max<<<egrid, eblk, 0, stream>>>(ei, E, Etot, a_s, a_d, m);
  gat_edge_denom<<<egrid, eblk, 0, stream>>>(ei, E, Etot, a_s, a_d, m, denom);

  gat_scatter<<<Etot, 128, 0, stream>>>(ei, E, a_s, a_d, m, denom, h, agg);

  gat_epilogue<<<nNodes, 128, 0, stream>>>(x, b, gamma, beta, agg,
                                           (float*)d_out);
}